// STGNNModel_7069516169283
// MI455X (gfx1250) — compile-verified
//
#include <hip/hip_runtime.h>
#include <math.h>

// ---------------- constants (match reference) ----------------
constexpr int cB  = 4;
constexpr int cT  = 12;
constexpr int cN  = 2000;
constexpr int cE  = 32000;
constexpr int cND = 16;
constexpr int cED = 8;
constexpr int cH  = 4;
constexpr int cD  = 16;
constexpr int cG  = 64;     // H*D
constexpr int cGH = 64;
constexpr int cBT = cB * cT;      // 48
constexpr int cBN = cB * cN;      // 8000
constexpr float LN_EPS = 1e-5f;

typedef float v2f __attribute__((ext_vector_type(2)));
typedef float v8f __attribute__((ext_vector_type(8)));

__device__ __forceinline__ v8f wmma4(v2f a, v2f b, v8f c) {
  // D(16x16,f32) += A(16x4,f32) * B(4x16,f32)  -- exact fp32 path on CDNA5
  return __builtin_amdgcn_wmma_f32_16x16x4_f32(false, a, false, b, (short)0, c,
                                               false, false);
}

// monotone float<->uint encoding so atomicMax(uint) == float max
__device__ __forceinline__ unsigned encF(float f) {
  unsigned u = __float_as_uint(f);
  return (u & 0x80000000u) ? ~u : (u | 0x80000000u);
}
__device__ __forceinline__ float decF(unsigned u) {
  return __uint_as_float((u & 0x80000000u) ? (u ^ 0x80000000u) : ~u);
}

// ================= 1. node projection: h = x @ W_node (WMMA) =================
// grid = (N/16, BT), block = 32 (one wave per 16x64 output tile)
__global__ void k_node_linear(const float* __restrict__ x,
                              const float* __restrict__ Wn,
                              float* __restrict__ hout) {
  const int lane = threadIdx.x;
  const int half = lane >> 4;
  const int lr   = lane & 15;
  const int m0   = blockIdx.x * 16;
  const int bt   = blockIdx.y;

  const float* xrow = x + ((size_t)bt * cN + (m0 + lr)) * cND;

  // A fragments for the 4 k-steps of K=16
  v2f a[4];
#pragma unroll
  for (int ks = 0; ks < 4; ++ks) {
    int k = ks * 4 + 2 * half;
    a[ks].x = xrow[k];
    a[ks].y = xrow[k + 1];
  }

  float* hbase = hout + (size_t)bt * cN * cG;
#pragma unroll
  for (int ct = 0; ct < 4; ++ct) {     // 4 column tiles of 16 (== heads)
    v8f c = {};
#pragma unroll
    for (int ks = 0; ks < 4; ++ks) {
      int k = ks * 4 + 2 * half;
      v2f b;
      b.x = Wn[(size_t)k * cG + ct * 16 + lr];
      b.y = Wn[(size_t)(k + 1) * cG + ct * 16 + lr];
      c = wmma4(a[ks], b, c);
    }
#pragma unroll
    for (int r = 0; r < 8; ++r) {
      int m = m0 + r + 8 * half;
      hbase[(size_t)m * cG + ct * 16 + lr] = c[r];
    }
  }
}

// ============ 2. per-node head scores + segment-accumulator init ============
__global__ void k_node_scores(const float* __restrict__ hlin,
                              const float* __restrict__ att,
                              float* __restrict__ sd, float* __restrict__ ss,
                              unsigned* __restrict__ smax,
                              float* __restrict__ ssum) {
  int idx = blockIdx.x * blockDim.x + threadIdx.x;
  if (idx >= cBT * cN) return;
  const float* row = hlin + (size_t)idx * cG;
  float ad[cH] = {0, 0, 0, 0}, as[cH] = {0, 0, 0, 0};
#pragma unroll
  for (int g = 0; g < cG; ++g) {
    int h = g >> 4, d = g & 15;
    float v = row[g];
    ad[h] += v * att[h * (3 * cD) + d];          // a_dst
    as[h] += v * att[h * (3 * cD) + cD + d];     // a_src
  }
#pragma unroll
  for (int h = 0; h < cH; ++h) {
    sd[(size_t)idx * cH + h]   = ad[h];
    ss[(size_t)idx * cH + h]   = as[h];
    smax[(size_t)idx * cH + h] = 0u;   // encF of any finite value > 0
    ssum[(size_t)idx * cH + h] = 0.f;
  }
}

// ===== 3. edge pass A: scores + segment max (edge proj recomputed, LDS W) ====
__global__ void k_edge_score(const float* __restrict__ ef,
                             const float* __restrict__ We,
                             const float* __restrict__ att,
                             const float* __restrict__ sd,
                             const float* __restrict__ ss,
                             const int* __restrict__ eidx,
                             float* __restrict__ score,
                             unsigned* __restrict__ smax) {
  __shared__ float sW[cED * cG];   // 512 floats
  __shared__ float sAe[cG];
  int tid = threadIdx.x;
  for (int i = tid; i < cED * cG; i += blockDim.x) sW[i] = We[i];
  for (int i = tid; i < cG; i += blockDim.x) {
    int h = i >> 4, d = i & 15;
    sAe[i] = att[h * (3 * cD) + 2 * cD + d];     // a_e
  }
  __syncthreads();

  int idx = blockIdx.x * blockDim.x + tid;
  if (idx >= cBT * cE) return;
  int bt = idx / cE, e = idx - bt * cE;
  int src = eidx[e], dst = eidx[cE + e];

  float ea[cED];
#pragma unroll
  for (int k = 0; k < cED; ++k) ea[k] = ef[((size_t)bt * cE + e) * cED + k];

  float se[cH] = {0, 0, 0, 0};
#pragma unroll 4
  for (int g = 0; g < cG; ++g) {
    float er = 0.f;
#pragma unroll
    for (int k = 0; k < cED; ++k) er += ea[k] * sW[k * cG + g];
    se[g >> 4] += er * sAe[g];
  }

  const float* sdr = sd + ((size_t)bt * cN + dst) * cH;
  const float* ssr = ss + ((size_t)bt * cN + src) * cH;
#pragma unroll
  for (int h = 0; h < cH; ++h) {
    float sc = sdr[h] + ssr[h] + se[h];
    sc = (sc >= 0.f) ? sc : 0.2f * sc;           // leaky_relu(0.2)
    score[(size_t)idx * cH + h] = sc;
    atomicMax(&smax[((size_t)bt * cN + dst) * cH + h], encF(sc));
  }
}

// ============== 4. edge pass B: exp(score - max) + segment sum ==============
__global__ void k_edge_expsum(const int* __restrict__ eidx,
                              float* __restrict__ score,
                              const unsigned* __restrict__ smax,
                              float* __restrict__ ssum) {
  int idx = blockIdx.x * blockDim.x + threadIdx.x;
  if (idx >= cBT * cE) return;
  int bt = idx / cE, e = idx - bt * cE;
  int dst = eidx[cE + e];
#pragma unroll
  for (int h = 0; h < cH; ++h) {
    float mx = decF(smax[((size_t)bt * cN + dst) * cH + h]);
    float ex = expf(score[(size_t)idx * cH + h] - mx);
    score[(size_t)idx * cH + h] = ex;            // now holds exp_s
    atomicAdd(&ssum[((size_t)bt * cN + dst) * cH + h], ex);
  }
}

// == 5. edge pass C: alpha, attn_stack accumulation, message scatter-add ====
__global__ void k_edge_agg(const float* __restrict__ ef,
                           const float* __restrict__ We,
                           const int* __restrict__ eidx,
                           const float* __restrict__ exbuf,
                           const float* __restrict__ ssum,
                           const float* __restrict__ hlin,
                           float* __restrict__ agg,
                           float* __restrict__ attn) {
  __shared__ float sW[cED * cG];
  int tid = threadIdx.x;
  for (int i = tid; i < cED * cG; i += blockDim.x) sW[i] = We[i];
  __syncthreads();

  int idx = blockIdx.x * blockDim.x + tid;
  if (idx >= cBT * cE) return;
  int bt = idx / cE, e = idx - bt * cE;
  int t = bt % cT;
  int src = eidx[e], dst = eidx[cE + e];

  float alpha[cH];
#pragma unroll
  for (int h = 0; h < cH; ++h) {
    float s = ssum[((size_t)bt * cN + dst) * cH + h];
    alpha[h] = exbuf[(size_t)idx * cH + h] / (s + 1e-16f);
    atomicAdd(&attn[((size_t)t * cE + e) * cH + h], alpha[h] * (1.0f / cB));
  }

  float ea[cED];
#pragma unroll
  for (int k = 0; k < cED; ++k) ea[k] = ef[((size_t)bt * cE + e) * cED + k];

  const float* hsrc = hlin + ((size_t)bt * cN + src) * cG;
  float* adst = agg + ((size_t)bt * cN + dst) * cG;
#pragma unroll 4
  for (int g = 0; g < cG; ++g) {
    float er = 0.f;
#pragma unroll
    for (int k = 0; k < cED; ++k) er += ea[k] * sW[k * cG + g];
    atomicAdd(&adst[g], alpha[g >> 4] * (hsrc[g] + er));
  }
}

// ==================== 6. ELU + LayerNorm (in place) =========================
__global__ void k_elu_ln(float* __restrict__ agg,
                         const float* __restrict__ gamma,
                         const float* __restrict__ beta) {
  int idx = blockIdx.x * blockDim.x + threadIdx.x;
  if (idx >= cBT * cN) return;
  float* row = agg + (size_t)idx * cG;
  float v[cG];
  float mu = 0.f;
#pragma unroll
  for (int g = 0; g < cG; ++g) {
    float x = row[g];
    x = (x > 0.f) ? x : (expf(x) - 1.f);         // elu(alpha=1)
    v[g] = x;
    mu += x;
  }
  mu *= (1.f / cG);
  float var = 0.f;
#pragma unroll
  for (int g = 0; g < cG; ++g) { float d = v[g] - mu; var += d * d; }
  var *= (1.f / cG);
  float inv = rsqrtf(var + LN_EPS);
#pragma unroll
  for (int g = 0; g < cG; ++g)
    row[g] = (v[g] - mu) * inv * gamma[g] + beta[g];
}

// =========== 7. fused GRU step: gi & gh GEMMs (WMMA) + gates ================
// grid = (BN/16, 4), block = 32. Each wave: 16 rows x one 16-col slab of H,
// computing r/z/n column tiles of both x@W_ih^T and h@W_hh^T (96 WMMAs).
__global__ void k_gru_step(const float* __restrict__ spatial,
                           const float* __restrict__ Wih,
                           const float* __restrict__ Whh,
                           const float* __restrict__ bih,
                           const float* __restrict__ bhh,
                           const float* __restrict__ hprev,
                           float* __restrict__ hnext, int t) {
  const int lane = threadIdx.x;
  const int half = lane >> 4;
  const int lr   = lane & 15;
  const int bn0  = blockIdx.x * 16;
  const int jg   = blockIdx.y;                   // col slab: jg*16 .. +15
  const int b    = bn0 / cN;                     // N % 16 == 0: tile in one b
  const int n0   = bn0 - b * cN;
  const int col  = jg * 16 + lr;

  const float* xrow = spatial + (((size_t)(b * cT + t) * cN) + (n0 + lr)) * cG;
  const float* hrow = hprev + (size_t)(bn0 + lr) * cGH;

  v8f ci0 = {}, ci1 = {}, ci2 = {}, ch0 = {}, ch1 = {}, ch2 = {};
#pragma unroll
  for (int ks = 0; ks < 16; ++ks) {              // K = 64
    int k = ks * 4 + 2 * half;
    v2f ax, ah;
    ax.x = xrow[k]; ax.y = xrow[k + 1];
    ah.x = hrow[k]; ah.y = hrow[k + 1];
    v2f bi0, bi1, bi2, bh0, bh1, bh2;
    bi0.x = Wih[(size_t)(col)*cG + k];        bi0.y = Wih[(size_t)(col)*cG + k + 1];
    bi1.x = Wih[(size_t)(col + 64) * cG + k]; bi1.y = Wih[(size_t)(col + 64) * cG + k + 1];
    bi2.x = Wih[(size_t)(col + 128) * cG + k];bi2.y = Wih[(size_t)(col + 128) * cG + k + 1];
    bh0.x = Whh[(size_t)(col)*cGH + k];        bh0.y = Whh[(size_t)(col)*cGH + k + 1];
    bh1.x = Whh[(size_t)(col + 64) * cGH + k]; bh1.y = Whh[(size_t)(col + 64) * cGH + k + 1];
    bh2.x = Whh[(size_t)(col + 128) * cGH + k];bh2.y = Whh[(size_t)(col + 128) * cGH + k + 1];
    ci0 = wmma4(ax, bi0, ci0);
    ci1 = wmma4(ax, bi1, ci1);
    ci2 = wmma4(ax, bi2, ci2);
    ch0 = wmma4(ah, bh0, ch0);
    ch1 = wmma4(ah, bh1, ch1);
    ch2 = wmma4(ah, bh2, ch2);
  }

  const float bir = bih[col], biz = bih[64 + col], bin = bih[128 + col];
  const float bhr = bhh[col], bhz = bhh[64 + col], bhn = bhh[128 + col];
#pragma unroll
  for (int r = 0; r < 8; ++r) {
    int m = bn0 + r + 8 * half;
    float gr = (ci0[r] + bir) + (ch0[r] + bhr);
    float gz = (ci1[r] + biz) + (ch1[r] + bhz);
    float rg = 1.f / (1.f + expf(-gr));
    float zg = 1.f / (1.f + expf(-gz));
    float ng = tanhf((ci2[r] + bin) + rg * (ch2[r] + bhn));
    float hp = hprev[(size_t)m * cGH + col];
    hnext[(size_t)m * cGH + col] = (1.f - zg) * ng + zg * hp;
  }
}

// ======================= 8. output head + h_i copy ==========================
__global__ void k_output(const float* __restrict__ hfin,
                         const float* __restrict__ Wout,
                         const float* __restrict__ bout,
                         float* __restrict__ out) {
  int bn = blockIdx.x * blockDim.x + threadIdx.x;
  if (bn >= cBN) return;
  const float* row = hfin + (size_t)bn * cGH;
  float acc = bout[0];
#pragma unroll
  for (int k = 0; k < cGH; ++k) {
    float v = row[k];
    acc += v * Wout[k];
    out[cBN + (size_t)bn * cGH + k] = v;         // h_i region
  }
  out[bn] = acc;                                 // pred region
}

// ============================================================================
extern "C" void kernel_launch(void* const* d_in, const int* in_sizes, int n_in,
                              void* d_out, int out_size, void* d_ws,
                              size_t ws_size, hipStream_t stream) {
  const float* node_f = (const float*)d_in[0];
  const float* edge_f = (const float*)d_in[1];
  const float* W_node = (const float*)d_in[2];
  const float* W_edge = (const float*)d_in[3];
  const float* att    = (const float*)d_in[4];
  const float* ln_g   = (const float*)d_in[5];
  const float* ln_b   = (const float*)d_in[6];
  const float* W_ih   = (const float*)d_in[7];
  const float* W_hh   = (const float*)d_in[8];
  const float* b_ih   = (const float*)d_in[9];
  const float* b_hh   = (const float*)d_in[10];
  const float* W_out  = (const float*)d_in[11];
  const float* b_out  = (const float*)d_in[12];
  const int*   eidx   = (const int*)d_in[13];

  float* ws = (float*)d_ws;
  // workspace layout (floats)
  const size_t off_h    = 0;                           // BT*N*G
  const size_t off_sd   = off_h + (size_t)cBT * cN * cG;
  const size_t off_ss   = off_sd + (size_t)cBT * cN * cH;
  const size_t off_smax = off_ss + (size_t)cBT * cN * cH;   // as uint
  const size_t off_ssum = off_smax + (size_t)cBT * cN * cH;
  const size_t off_sc   = off_ssum + (size_t)cBT * cN * cH; // BT*E*H
  const size_t off_agg  = off_sc + (size_t)cBT * cE * cH;   // BT*N*G
  const size_t off_h0   = off_agg + (size_t)cBT * cN * cG;  // BN*GH
  const size_t off_h1   = off_h0 + (size_t)cBN * cGH;

  float* hlin  = ws + off_h;
  float* sd    = ws + off_sd;
  float* ssb   = ws + off_ss;
  unsigned* smax = (unsigned*)(ws + off_smax);
  float* ssum  = ws + off_ssum;
  float* score = ws + off_sc;
  float* agg   = ws + off_agg;
  float* hb[2] = {ws + off_h0, ws + off_h1};

  float* out      = (float*)d_out;
  float* attn_out = out + cBN + (size_t)cBN * cGH;     // T*E*H region

  // zero-init accumulators (capture-safe async memsets)
  hipMemsetAsync(agg, 0, (size_t)cBT * cN * cG * sizeof(float), stream);
  hipMemsetAsync(hb[0], 0, (size_t)cBN * cGH * sizeof(float), stream);
  hipMemsetAsync(attn_out, 0, (size_t)cT * cE * cH * sizeof(float), stream);

  // GAT
  k_node_linear<<<dim3(cN / 16, cBT), 32, 0, stream>>>(node_f, W_node, hlin);
  k_node_scores<<<(cBT * cN + 255) / 256, 256, 0, stream>>>(hlin, att, sd, ssb,
                                                            smax, ssum);
  const int eblocks = (cBT * cE + 255) / 256;
  k_edge_score<<<eblocks, 256, 0, stream>>>(edge_f, W_edge, att, sd, ssb, eidx,
                                            score, smax);
  k_edge_expsum<<<eblocks, 256, 0, stream>>>(eidx, score, smax, ssum);
  k_edge_agg<<<eblocks, 256, 0, stream>>>(edge_f, W_edge, eidx, score, ssum,
                                          hlin, agg, attn_out);
  k_elu_ln<<<(cBT * cN + 255) / 256, 256, 0, stream>>>(agg, ln_g, ln_b);

  // GRU scan over T (ping-pong; T even -> final state lands in hb[0])
  for (int t = 0; t < cT; ++t) {
    k_gru_step<<<dim3(cBN / 16, 4), 32, 0, stream>>>(
        agg, W_ih, W_hh, b_ih, b_hh, hb[t & 1], hb[(t + 1) & 1], t);
  }

  // head
  k_output<<<(cBN + 255) / 256, 256, 0, stream>>>(hb[0], W_out, b_out, out);
}